// Atomistic_49263274885346
// MI455X (gfx1250) — compile-verified
//
#include <hip/hip_runtime.h>

// Per-atom energy: e[n] = dot(x[n,:], W[species[n],:]) + b[species[n]]
// then segment-sum over sorted struct_idx -> out[10000].
//
// Bandwidth-bound: 512MB of x @ 23.3 TB/s ~= 22us floor. Dot products via
// V_WMMA_F32_16X16X4_F32: per wave, 16 atoms x 128 feats against a 16x128
// B panel whose rows 0..3 are W and rows 4..15 are zero (pre-padded in LDS
// so the ds_load feeds the WMMA directly, no per-iteration masking).

typedef __attribute__((ext_vector_type(2))) float v2f;
typedef __attribute__((ext_vector_type(8))) float v8f;

#define D_FEAT 128
#define N_SPECIES 4
#define B_ROWS 16          // padded species rows for the WMMA N dimension

__global__ void zero_kernel(float* __restrict__ out, int n) {
    int i = blockIdx.x * blockDim.x + threadIdx.x;
    if (i < n) out[i] = 0.0f;
}

__global__ __launch_bounds__(256) void atom_energy_kernel(
    const float* __restrict__ x,
    const float* __restrict__ W,
    const float* __restrict__ b,
    const int*   __restrict__ species,
    const int*   __restrict__ sidx,
    float*       __restrict__ out,
    int nAtoms)
{
    __shared__ float Bpanel[B_ROWS * D_FEAT];    // 8 KB: rows 0..3 = W, 4..15 = 0
    __shared__ float Dtile[8][256];              // one 16x16 f32 tile per wave

    // Stage zero-padded B panel once per block.
    for (int t = threadIdx.x; t < B_ROWS * D_FEAT; t += blockDim.x)
        Bpanel[t] = (t < N_SPECIES * D_FEAT) ? W[t] : 0.0f;
    __syncthreads();

    const int lane        = threadIdx.x & 31;
    const int waveInBlock = threadIdx.x >> 5;
    const int waveGlobal  = blockIdx.x * (blockDim.x >> 5) + waveInBlock;
    const int base        = waveGlobal * 16;     // 16 atoms per wave

    const int hi    = lane >> 4;                 // 0: K={0,1}, 1: K={2,3}
    const int lo    = lane & 15;
    const int khalf = hi * 2;

    // A-operand row (atom) for this lane, clamped for the tail tile.
    const int    arow = min(base + lo, nAtoms - 1);
    const float* xrow = x + (size_t)arow * D_FEAT;

    // B-operand column for this lane: padded rows make this unconditional.
    const float* wrow = &Bpanel[lo * D_FEAT];

    // D = sum_k A(16x4) * B(4x16): 32 x V_WMMA_F32_16X16X4_F32.
    v8f acc = {};
    #pragma unroll 4
    for (int k = 0; k < D_FEAT; k += 4) {
        v2f a  = *(const v2f*)(xrow + k + khalf);   // global_load_b64
        v2f bb = *(const v2f*)(wrow + k + khalf);   // ds_load_b64, feeds wmma
        acc = __builtin_amdgcn_wmma_f32_16x16x4_f32(
            /*neg_a=*/false, a, /*neg_b=*/false, bb,
            /*c_mod=*/(short)0, acc, /*reuse_a=*/false, /*reuse_b=*/false);
    }

    // Spill D tile to this wave's LDS region: D(M,N), M = v + 8*hi, N = lo.
    float* dt = &Dtile[waveInBlock][0];
    #pragma unroll
    for (int v = 0; v < 8; ++v) {
        dt[(v + 8 * hi) * 16 + lo] = acc[v];
    }
    __builtin_amdgcn_wave_barrier();  // per-wave DS ordering; pin the schedule

    // Lanes 0..15: pick this atom's species column, add bias.
    float val = 0.0f;
    int   sid = -1;
    if (lane < 16) {
        int atom = base + lane;
        if (atom < nAtoms) {
            int sp = species[atom];
            val = dt[lane * 16 + sp] + b[sp];
            sid = sidx[atom];
        }
    }

    // Segmented inclusive scan over the (sorted) struct ids within the tile,
    // so only the last lane of each run issues a global fp32 atomic.
    #pragma unroll
    for (int off = 1; off < 16; off <<= 1) {
        float pv = __shfl_up(val, off, 32);
        int   ps = __shfl_up(sid, off, 32);
        if (lane >= off && ps == sid) val += pv;
    }
    int  nxt  = __shfl_down(sid, 1, 32);
    bool last = (lane == 15) || (nxt != sid);
    if (lane < 16 && sid >= 0 && last) {
        __hip_atomic_fetch_add(&out[sid], val,
                               __ATOMIC_RELAXED, __HIP_MEMORY_SCOPE_AGENT);
    }
}

extern "C" void kernel_launch(void* const* d_in, const int* in_sizes, int n_in,
                              void* d_out, int out_size, void* d_ws, size_t ws_size,
                              hipStream_t stream) {
    const float* x       = (const float*)d_in[0];
    const float* W       = (const float*)d_in[1];
    const float* b       = (const float*)d_in[2];
    const int*   species = (const int*)d_in[3];
    const int*   sidx    = (const int*)d_in[4];
    float*       out     = (float*)d_out;

    const int nAtoms = in_sizes[3];  // species has one entry per atom

    // Harness poisons d_out; zero it each call (deterministic under graph replay).
    zero_kernel<<<(out_size + 255) / 256, 256, 0, stream>>>(out, out_size);

    const int wavesNeeded   = (nAtoms + 15) / 16;  // 16 atoms per wave
    const int wavesPerBlock = 256 / 32;
    const int blocks        = (wavesNeeded + wavesPerBlock - 1) / wavesPerBlock;
    atom_energy_kernel<<<blocks, 256, 0, stream>>>(x, W, b, species, sidx, out, nAtoms);
}